// EikonalFieldProvider_10033043603755
// MI455X (gfx1250) — compile-verified
//
#include <hip/hip_runtime.h>

#define NPSI 256
#define MM 8
#define KK 4

__device__ __forceinline__ uint32_t lds_off_u32(const void* p) {
  // flat shared address: addr[31:0] == workgroup-relative LDS byte offset (CDNA5 aperture rule)
  return (uint32_t)(uintptr_t)p;
}

__global__ __launch_bounds__(256)
void eikonal_field_kernel(const float* __restrict__ tptr,
                          const float* __restrict__ psi,
                          const float* __restrict__ theta,
                          const float* __restrict__ varphi,
                          const float* __restrict__ q_vals,
                          const float* __restrict__ aoff_vals,
                          const float* __restrict__ theta0,
                          const float* __restrict__ omega,
                          const float* __restrict__ psi0,
                          const float* __restrict__ psi_scale,
                          const float* __restrict__ alpha_scale,
                          const float* __restrict__ gh,      // [M][4][6] flat
                          const int*   __restrict__ nmode,
                          float* __restrict__ out_phi,
                          float* __restrict__ out_apar,
                          int N)
{
  // ---- Stage q/aoff tables into LDS via CDNA5 async global->LDS DMA ----
  __shared__ float4 tab[NPSI];   // {q, aoff, dq, daoff} per knot, 4 KB
  const int t = threadIdx.x;     // blockDim.x == 256 == NPSI
  {
    uint32_t lq = lds_off_u32(&tab[t].x);
    uint32_t la = lds_off_u32(&tab[t].y);
    uint64_t gq = (uint64_t)(uintptr_t)(q_vals + t);
    uint64_t ga = (uint64_t)(uintptr_t)(aoff_vals + t);
    asm volatile("global_load_async_to_lds_b32 %0, %1, off"
                 :: "v"(lq), "v"(gq) : "memory");
    asm volatile("global_load_async_to_lds_b32 %0, %1, off"
                 :: "v"(la), "v"(ga) : "memory");
    asm volatile("s_wait_asynccnt 0x0" ::: "memory");
  }
  __syncthreads();

  // ---- FD gradients (jnp.gradient semantics, unit-index spacing; the H's cancel) ----
  {
    int im = (t > 0)        ? t - 1 : 0;
    int ip = (t < NPSI - 1) ? t + 1 : NPSI - 1;
    float sc = (t == 0 || t == NPSI - 1) ? 1.0f : 0.5f;
    float dq = (tab[ip].x - tab[im].x) * sc;   // reads .x/.y, writes .z/.w: disjoint DWORDs
    float da = (tab[ip].y - tab[im].y) * sc;
    tab[t].z = dq;
    tab[t].w = da;
  }
  __syncthreads();

  const float tval = tptr[0];
  const int stride = gridDim.x * blockDim.x;
  const float L2E_HALF = -0.7213475204444817f;     // -0.5*log2(e)
  const float INV_2PI  =  0.15915494309189535f;
  const float TWO_PI   =  6.2831853071795864f;

  for (int idx = blockIdx.x * blockDim.x + t; idx < N; idx += stride) {
    const float ps = psi[idx];
    const float th = theta[idx];
    const float vp = varphi[idx];

    // ---- cubic Hermite interp of (q, aoff) at ps; one ds_load_b128 per knot ----
    float u  = ps * (float)(NPSI - 1);
    float fi = floorf(u);
    fi = fminf(fmaxf(fi, 0.0f), (float)(NPSI - 2));
    const int i0 = (int)fi;
    const float s  = u - fi;
    const float4 A = tab[i0];
    const float4 B = tab[i0 + 1];
    const float s2 = s * s, s3 = s2 * s;
    const float w0 = 2.f * s3 - 3.f * s2 + 1.f;
    const float w1 = s3 - 2.f * s2 + s;           // multiplies raw index-gradient (H*d)
    const float w2 = 3.f * s2 - 2.f * s3;
    const float w3 = s3 - s2;
    const float qv = w0 * A.x + w1 * A.z + w2 * B.x + w3 * B.z;
    const float av = w0 * A.y + w1 * A.w + w2 * B.y + w3 * B.w;

    // alpha_k = q*(theta + 2*pi*k) + aoff
    float alpha_k[KK];
#pragma unroll
    for (int k = 0; k < KK; ++k)
      alpha_k[k] = fmaf(qv, th + TWO_PI * (float)k, av);

    float acc_phi = 0.f, acc_apar = 0.f;

#pragma unroll 2
    for (int m = 0; m < MM; ++m) {
      // wave-uniform -> s_load through scalar cache
      const float th0    = theta0[m];
      const float om     = omega[m];
      const float p0m    = psi0[m];
      const float inv_ps = __builtin_amdgcn_rcpf(psi_scale[m]);
      const float inv_as = __builtin_amdgcn_rcpf(alpha_scale[m]);
      const float nf     = (float)nmode[m];
      const float* c = gh + m * 24;

      const float x   = (ps - p0m) * inv_ps;
      const float x2  = x * x;
      const float h20 = 4.f * x2 - 2.f;
      const float ex  = __builtin_amdgcn_exp2f(x2 * L2E_HALF);
      // alpha0 via linearity of the cubic interp: interp(q*th0 + aoff) = qv*th0 + av
      const float a0   = fmaf(qv, th0, av);
      const float base = fmaf(om, tval, -nf * vp);  // omega*t - n*varphi

#pragma unroll
      for (int k = 0; k < KK; ++k) {
        const float da  = alpha_k[k] - a0;          // == y * alpha_scale
        const float y   = da * inv_as;
        const float y2  = y * y;
        const float g   = ex * __builtin_amdgcn_exp2f(y2 * L2E_HALF);
        const float h11 = x * y;
        const float h02 = 4.f * y2 - 2.f;

        const float P0 = fmaf(c[5],  h02, fmaf(c[4],  h11, fmaf(c[3],  h20,
                          fmaf(c[2],  y, fmaf(c[1],  x, c[0])))));
        const float P1 = fmaf(c[11], h02, fmaf(c[10], h11, fmaf(c[9],  h20,
                          fmaf(c[8],  y, fmaf(c[7],  x, c[6])))));
        const float P2 = fmaf(c[17], h02, fmaf(c[16], h11, fmaf(c[15], h20,
                          fmaf(c[14], y, fmaf(c[13], x, c[12])))));
        const float P3 = fmaf(c[23], h02, fmaf(c[22], h11, fmaf(c[21], h20,
                          fmaf(c[20], y, fmaf(c[19], x, c[18])))));

        // phase = base + n*(alpha_k - alpha0); v_sin/v_cos take cycles
        const float cyc = fmaf(nf, da, base) * INV_2PI;
        const float sp  = __builtin_amdgcn_sinf(cyc);
        const float cp  = __builtin_amdgcn_cosf(cyc);

        acc_phi  = fmaf(g, fmaf(-P1, sp, P0 * cp), acc_phi);
        acc_apar = fmaf(g, fmaf(-P3, sp, P2 * cp), acc_apar);
      }
    }

    out_phi[idx]  = acc_phi;
    out_apar[idx] = acc_apar;
  }
}

extern "C" void kernel_launch(void* const* d_in, const int* in_sizes, int n_in,
                              void* d_out, int out_size, void* d_ws, size_t ws_size,
                              hipStream_t stream) {
  (void)n_in; (void)d_ws; (void)ws_size; (void)out_size;
  const float* tptr        = (const float*)d_in[0];
  const float* psi         = (const float*)d_in[1];
  const float* theta       = (const float*)d_in[2];
  const float* varphi      = (const float*)d_in[3];
  const float* q_vals      = (const float*)d_in[4];
  const float* aoff_vals   = (const float*)d_in[5];
  const float* theta0      = (const float*)d_in[6];
  const float* omega       = (const float*)d_in[7];
  const float* psi0        = (const float*)d_in[8];
  const float* psi_scale   = (const float*)d_in[9];
  const float* alpha_scale = (const float*)d_in[10];
  const float* gh          = (const float*)d_in[11];
  const int*   nmode       = (const int*)d_in[12];

  const int N = in_sizes[1];
  float* out_phi  = (float*)d_out;
  float* out_apar = out_phi + N;

  const int threads = 256;                 // must equal NPSI for table staging
  const int blocks  = (N + threads - 1) / threads;
  eikonal_field_kernel<<<blocks, threads, 0, stream>>>(
      tptr, psi, theta, varphi, q_vals, aoff_vals, theta0, omega,
      psi0, psi_scale, alpha_scale, gh, nmode, out_phi, out_apar, N);
}